// BWSGODE_naive_int_70188355551732
// MI455X (gfx1250) — compile-verified
//
#include <hip/hip_runtime.h>
#include <cstdint>

// BWSG ODE: strictly sequential 4-state recurrence, 8192 steps, emit every row.
// Latency-bound: one wave32, all lanes compute the recurrence redundantly
// (zero divergence in the chain). Output rows are staged in LDS and drained to
// HBM with CDNA5 async LDS->global stores (ASYNCcnt) so stores never stall the
// dependent VALU chain. Double-buffered 32-row chunks.

#define CHUNK 32          // rows per LDS buffer
#define ROWF  5           // floats per row
#define BUFSZ (CHUNK * ROWF)

__global__ __launch_bounds__(32)
void bwsg_ode_kernel(const float* __restrict__ y0,
                     const float* __restrict__ params,
                     const int*   __restrict__ nsteps_p,
                     float* __restrict__ out)
{
    __shared__ float smem[2 * BUFSZ];

    const int lane = threadIdx.x;
    const int n    = nsteps_p[0];

    const float i_val          = y0[4];
    const float interventional = (i_val != 0.0f) ? 1.0f : 0.0f;
    const float one_minus      = 1.0f - interventional;
    const float thr            = 5.0f + i_val - 1.0f;   // (5.0 + i) - 1.0, ref order

    const float p0 = params[0], p1 = params[1], p2 = params[2], p3 = params[3];
    const float p4 = params[4], p5 = params[5], p6 = params[6], p7 = params[7];
    const float p8 = params[8], p9 = params[9];

    float B = y0[0], W = y0[1], S = y0[2], G = y0[3];

#define ODE_STEP(RJ)                                                          \
    do {                                                                      \
        float j_    = (float)(RJ);                                            \
        float mask_ = interventional * ((j_ >= thr) ? 1.0f : 0.0f) + one_minus;\
        float Bm_   = B * mask_;                                              \
        float dG_   = (p0 * (1.0f - G) - p1 * S) * G;                         \
        float dS_   = S * (p2 * G - p3 * (W + Bm_) - p4);                     \
        float dW_   = W * (p5 * S - p6 * Bm_ - p7) * W;                       \
        float dB_   = Bm_ * (p8 * (S + W) - p9);                              \
        B += dB_; W += dW_; S += dS_; G += dG_;                               \
    } while (0)

    int r = 0;
    const int full = n / CHUNK;

    for (int c = 0; c < full; ++c) {
        const int buf = c & 1;

        // Buffer `buf` was last async-drained 2 chunks ago (5 stores/chunk).
        // ASYNCcnt <= 5 guarantees those 5 completed (async stores retire in
        // order); the <=5 still in flight belong to the other buffer.
        asm volatile("s_wait_asynccnt 5" ::: "memory");

        #pragma unroll 4
        for (int k = 0; k < CHUNK; ++k, ++r) {
            if (r > 0) ODE_STEP(r);
            float v = (lane == 0) ? B
                    : (lane == 1) ? W
                    : (lane == 2) ? S
                    : (lane == 3) ? G
                    : i_val;
            if (lane < ROWF) smem[buf * BUFSZ + k * ROWF + lane] = v;
        }

        // LDS writes must land before the async DMA reads them (DScnt vs ASYNCcnt).
        asm volatile("s_wait_dscnt 0" ::: "memory");

        // Drain 160 dwords: 5 async stores x 32 lanes, fully coalesced.
        float* gbase = out + (size_t)c * BUFSZ;
        #pragma unroll
        for (int t = 0; t < ROWF; ++t) {
            uint64_t gaddr = (uint64_t)(uintptr_t)(gbase + t * 32 + lane);
            uint32_t laddr = (uint32_t)(uintptr_t)(&smem[buf * BUFSZ + t * 32 + lane]);
            asm volatile("global_store_async_from_lds_b32 %0, %1, off"
                         :: "v"(gaddr), "v"(laddr)
                         : "memory");
        }
        // S_ENDPGM performs an implicit wait-idle (all counters, incl. ASYNCcnt),
        // so the final chunks are guaranteed visible before kernel completion.
    }

    // Tail rows (n not a multiple of CHUNK): direct stores, lanes 0..4.
    for (; r < n; ++r) {
        if (r > 0) ODE_STEP(r);
        float v = (lane == 0) ? B
                : (lane == 1) ? W
                : (lane == 2) ? S
                : (lane == 3) ? G
                : i_val;
        if (lane < ROWF) out[(size_t)r * ROWF + lane] = v;
    }
#undef ODE_STEP
}

extern "C" void kernel_launch(void* const* d_in, const int* in_sizes, int n_in,
                              void* d_out, int out_size, void* d_ws, size_t ws_size,
                              hipStream_t stream) {
    (void)in_sizes; (void)n_in; (void)out_size; (void)d_ws; (void)ws_size;
    const float* y0     = (const float*)d_in[0];
    const float* params = (const float*)d_in[1];
    const int*   nsteps = (const int*)d_in[2];
    float*       out    = (float*)d_out;

    // Single wave32: the recurrence is inherently serial; more waves cannot help.
    bwsg_ode_kernel<<<dim3(1), dim3(32), 0, stream>>>(y0, params, nsteps, out);
}